// DocoderZeroHiddenDropConnect_50989851738709
// MI455X (gfx1250) — compile-verified
//
#include <hip/hip_runtime.h>
#include <hip/hip_bf16.h>
#include <math.h>

// Sizes from the reference
#define BSZ   128
#define PP    196
#define FF    2048
#define EE    512
#define HH    1024
#define VV    10000
#define TMAXX 30
#define NSTEP 31            // 1 init cell + 30 scan cells

typedef __attribute__((ext_vector_type(2))) float v2f;
typedef __attribute__((ext_vector_type(8))) float v8f;

__device__ __forceinline__ float sigf(float x) { return 1.0f / (1.0f + expf(-x)); }

// ---------------------------------------------------------------------------
// Core: one wave computes an (16*MT) x (16*NT) fp32 tile of A(MxK) @ Wt^T,
// Wt row-major (N x K) so B(k,n) = Wt[n*ldw + k], accumulating K in [k0,k1).
// V_WMMA_F32_16X16X4_F32: A frag = float2/lane (lanes 0-15: K=kb+0..1,
// lanes 16-31: K=kb+2..3); B frag mirrors it along N.
// Register-blocks MT x NT to reuse A and B fragments across WMMAs.
// ---------------------------------------------------------------------------
template <int MT, int NT>
__device__ __forceinline__ void wmma_gemm_f32(const float* __restrict__ A, int lda, int mtile0,
                                              const float* __restrict__ Wt, int ldw, int ntile0,
                                              int k0, int k1, int lane, v8f acc[MT][NT]) {
    const int koff = (lane >> 4) << 1;
    int mrow[MT], ncol[NT];
#pragma unroll
    for (int mi = 0; mi < MT; ++mi) mrow[mi] = (mtile0 + mi) * 16 + (lane & 15);
#pragma unroll
    for (int j = 0; j < NT; ++j) ncol[j] = (ntile0 + j) * 16 + (lane & 15);

#pragma unroll 4
    for (int kb = k0; kb < k1; kb += 4) {
        const int kk = kb + koff;
        v2f a[MT], b[NT];
#pragma unroll
        for (int mi = 0; mi < MT; ++mi) a[mi] = *(const v2f*)(A + (size_t)mrow[mi] * lda + kk);
#pragma unroll
        for (int j = 0; j < NT; ++j) b[j] = *(const v2f*)(Wt + (size_t)ncol[j] * ldw + kk);
#pragma unroll
        for (int mi = 0; mi < MT; ++mi)
#pragma unroll
            for (int j = 0; j < NT; ++j)
                acc[mi][j] = __builtin_amdgcn_wmma_f32_16x16x4_f32(
                    false, a[mi], false, b[j], (short)0, acc[mi][j], false, false);
    }
}

// ---------------------------------------------------------------------------
// 1) feat[b][f] = mean_p enc[b][p][f]
// ---------------------------------------------------------------------------
__global__ void mean_kernel(const float* __restrict__ enc, float* __restrict__ feat) {
    int gid = blockIdx.x * blockDim.x + threadIdx.x;          // BSZ*FF threads
    int b = gid >> 11, f = gid & (FF - 1);
    const float* p = enc + (size_t)b * PP * FF + f;
    float s = 0.0f;
#pragma unroll 4
    for (int i = 0; i < PP; ++i) s += p[(size_t)i * FF];
    feat[gid] = s * (1.0f / (float)PP);
}

// ---------------------------------------------------------------------------
// 2) x0 = BN(sigmoid(feat @ W_init^T + b_init))   (128x2048)@(2048x512)
// ---------------------------------------------------------------------------
__global__ void x0_kernel(const float* __restrict__ feat, const float* __restrict__ W_init,
                          const float* __restrict__ b_init, const float* __restrict__ gamma,
                          const float* __restrict__ beta, float* __restrict__ x0) {
    const int lane = threadIdx.x & 31;
    const int w = blockIdx.x * (blockDim.x >> 5) + (threadIdx.x >> 5);
    const int NGRP = EE / 64;                 // 8 groups of 4 ntiles
    const int mtile = w / NGRP, grp = w % NGRP;
    v8f acc[1][4] = {};
    wmma_gemm_f32<1, 4>(feat, FF, mtile, W_init, FF, grp * 4, 0, FF, lane, acc);
    const float inv = rsqrtf(1.0f + 1e-5f);
#pragma unroll
    for (int j = 0; j < 4; ++j) {
        const int n = (grp * 4 + j) * 16 + (lane & 15);
        const float bi = b_init[n], ga = gamma[n], be = beta[n];
#pragma unroll
        for (int r = 0; r < 8; ++r) {
            const int m = mtile * 16 + r + ((lane >> 4) << 3);
            const float v = sigf(acc[0][j][r] + bi);
            x0[m * EE + n] = ga * (v * inv) + be;
        }
    }
}

// ---------------------------------------------------------------------------
// 3) Build X (NSTEP x B x E): X[0]=x0, X[t]=embedding[gt[:, t-1]]
// ---------------------------------------------------------------------------
__global__ void gather_kernel(const float* __restrict__ x0, const float* __restrict__ emb,
                              const int* __restrict__ gt, float* __restrict__ X) {
    int gid = blockIdx.x * blockDim.x + threadIdx.x;          // NSTEP*BSZ*EE
    int k = gid & (EE - 1);
    int row = gid >> 9;                                       // t*128 + b
    int t = row >> 7, b = row & (BSZ - 1);
    float v;
    if (t == 0) v = x0[b * EE + k];
    else        v = emb[(size_t)gt[b * (TMAXX + 1) + (t - 1)] * EE + k];
    X[gid] = v;
}

// ---------------------------------------------------------------------------
// 4) G = X(3968x512) @ W_ih^T(512x4096) + (b_ih + b_hh)    MT=2 x NT=4
// ---------------------------------------------------------------------------
__global__ void gatesx_kernel(const float* __restrict__ X, const float* __restrict__ W_ih,
                              const float* __restrict__ b_ih, const float* __restrict__ b_hh,
                              float* __restrict__ G) {
    const int lane = threadIdx.x & 31;
    const int w = blockIdx.x * (blockDim.x >> 5) + (threadIdx.x >> 5);
    const int NGRP = (4 * HH) / 64;           // 64 n-groups
    const int mgrp = w / NGRP, grp = w % NGRP;
    v8f acc[2][4] = {};
    wmma_gemm_f32<2, 4>(X, EE, mgrp * 2, W_ih, EE, grp * 4, 0, EE, lane, acc);
#pragma unroll
    for (int j = 0; j < 4; ++j) {
        const int n = (grp * 4 + j) * 16 + (lane & 15);
        const float bb = b_ih[n] + b_hh[n];
#pragma unroll
        for (int mi = 0; mi < 2; ++mi)
#pragma unroll
            for (int r = 0; r < 8; ++r) {
                const int m = (mgrp * 2 + mi) * 16 + r + ((lane >> 4) << 3);
                G[(size_t)m * (4 * HH) + n] = acc[mi][j][r] + bb;
            }
    }
}

__global__ void zero_kernel(float* __restrict__ p) {
    p[blockIdx.x * blockDim.x + threadIdx.x] = 0.0f;
}

// ---------------------------------------------------------------------------
// 5a) gates_part[ks] = (ks==0 ? G[t] : 0) + h[:, ksK/2:(ks+1)K/2] @ W_hh^T
//     Split-K x NT=2 => 2048 waves/step to fill the machine during the
//     latency-bound recurrence (W_hh is L2-resident).
// ---------------------------------------------------------------------------
__global__ void gatesh_kernel(const float* __restrict__ h, const float* __restrict__ W_hh,
                              const float* __restrict__ G, int t, float* __restrict__ gates) {
    const int lane = threadIdx.x & 31;
    const int w = blockIdx.x * (blockDim.x >> 5) + (threadIdx.x >> 5);
    const int NGRP = (4 * HH) / 32;           // 128 n-groups (NT=2)
    const int mtile = w / NGRP, grp = w % NGRP;
    const int ks = blockIdx.y;                // split-K index (0 or 1)
    v8f acc[1][2] = {};
    if (ks == 0) {
#pragma unroll
        for (int j = 0; j < 2; ++j) {
            const int n = (grp * 2 + j) * 16 + (lane & 15);
#pragma unroll
            for (int r = 0; r < 8; ++r) {
                const int m = mtile * 16 + r + ((lane >> 4) << 3);
                acc[0][j][r] = G[((size_t)(t * BSZ + m)) * (4 * HH) + n];
            }
        }
    }
    wmma_gemm_f32<1, 2>(h, HH, mtile, W_hh, HH, grp * 2,
                        ks * (HH / 2), (ks + 1) * (HH / 2), lane, acc);
    float* gout = gates + (size_t)ks * BSZ * 4 * HH;
#pragma unroll
    for (int j = 0; j < 2; ++j) {
        const int n = (grp * 2 + j) * 16 + (lane & 15);
#pragma unroll
        for (int r = 0; r < 8; ++r) {
            const int m = mtile * 16 + r + ((lane >> 4) << 3);
            gout[(size_t)m * (4 * HH) + n] = acc[0][j][r];
        }
    }
}

// ---------------------------------------------------------------------------
// 5b) LSTM cell elementwise (folds the two split-K partials); h -> Hall (t>=1)
// ---------------------------------------------------------------------------
__global__ void lstm_update_kernel(const float* __restrict__ gates, float* __restrict__ h,
                                   float* __restrict__ c, float* __restrict__ Hall, int t) {
    int gid = blockIdx.x * blockDim.x + threadIdx.x;          // BSZ*HH
    int b = gid >> 10, j = gid & (HH - 1);
    const float* g0 = gates + (size_t)b * (4 * HH);
    const float* g1 = g0 + (size_t)BSZ * 4 * HH;
    float i_ = sigf(g0[j]          + g1[j]);
    float f_ = sigf(g0[HH + j]     + g1[HH + j]);
    float gg = tanhf(g0[2 * HH + j] + g1[2 * HH + j]);
    float o_ = sigf(g0[3 * HH + j] + g1[3 * HH + j]);
    float cn = f_ * c[gid] + i_ * gg;
    float hn = o_ * tanhf(cn);
    c[gid] = cn;
    h[gid] = hn;
    if (t >= 1) Hall[(size_t)((t - 1) * BSZ + b) * HH + j] = hn;
}

// ---------------------------------------------------------------------------
// 6) out = mask( Hall(3840x1024) @ W_fc^T(1024x10000) + b_fc )   MT=2 x NT=5
//    10000 = 625 ntiles; NT=5 groups kept for exact division: use NT=5, MT=2.
// ---------------------------------------------------------------------------
__global__ void fc_kernel(const float* __restrict__ Hall, const float* __restrict__ W_fc,
                          const float* __restrict__ b_fc, const int* __restrict__ clen,
                          float* __restrict__ out) {
    const int lane = threadIdx.x & 31;
    const int w = blockIdx.x * (blockDim.x >> 5) + (threadIdx.x >> 5);
    const int NGRP = VV / (16 * 5);           // 125 n-groups (NT=5)
    const int mgrp = w / NGRP, grp = w % NGRP;
    v8f acc[2][5] = {};
    wmma_gemm_f32<2, 5>(Hall, HH, mgrp * 2, W_fc, HH, grp * 5, 0, HH, lane, acc);
#pragma unroll
    for (int j = 0; j < 5; ++j) {
        const int n = (grp * 5 + j) * 16 + (lane & 15);
        const float bf = b_fc[n];
#pragma unroll
        for (int mi = 0; mi < 2; ++mi)
#pragma unroll
            for (int r = 0; r < 8; ++r) {
                const int row = (mgrp * 2 + mi) * 16 + r + ((lane >> 4) << 3); // t*128+b
                const int tt = row >> 7, b = row & (BSZ - 1);
                const float v = (tt < clen[b] - 1) ? (acc[mi][j][r] + bf) : 0.0f;
                out[(size_t)b * (TMAXX * VV) + (size_t)tt * VV + n] = v;
            }
    }
}

// ---------------------------------------------------------------------------
extern "C" void kernel_launch(void* const* d_in, const int* in_sizes, int n_in,
                              void* d_out, int out_size, void* d_ws, size_t ws_size,
                              hipStream_t stream) {
    const float* enc    = (const float*)d_in[0];
    const int*   gt     = (const int*)  d_in[1];
    const int*   clen   = (const int*)  d_in[2];
    const float* emb    = (const float*)d_in[3];
    const float* W_init = (const float*)d_in[4];
    const float* b_init = (const float*)d_in[5];
    const float* gamma  = (const float*)d_in[6];
    const float* beta   = (const float*)d_in[7];
    const float* W_ih   = (const float*)d_in[8];
    const float* b_ih   = (const float*)d_in[9];
    const float* W_hh   = (const float*)d_in[10];
    const float* b_hh   = (const float*)d_in[11];
    const float* W_fc   = (const float*)d_in[12];
    const float* b_fc   = (const float*)d_in[13];
    float* out = (float*)d_out;

    // Workspace layout (floats), total ~23.9M floats (~95 MB)
    float* ws    = (float*)d_ws;
    float* feat  = ws;                                    // 128*2048
    float* x0    = feat  + (size_t)BSZ * FF;              // 128*512
    float* X     = x0    + (size_t)BSZ * EE;              // 31*128*512
    float* G     = X     + (size_t)NSTEP * BSZ * EE;      // 31*128*4096
    float* Hall  = G     + (size_t)NSTEP * BSZ * 4 * HH;  // 30*128*1024
    float* gates = Hall  + (size_t)TMAXX * BSZ * HH;      // 2 * 128*4096 (split-K)
    float* hbuf  = gates + (size_t)2 * BSZ * 4 * HH;      // 128*1024
    float* cbuf  = hbuf  + (size_t)BSZ * HH;              // 128*1024 (contiguous w/ hbuf)

    mean_kernel  <<<(BSZ * FF) / 256, 256, 0, stream>>>(enc, feat);
    x0_kernel    <<<(8 * (EE / 64)) / 8, 256, 0, stream>>>(feat, W_init, b_init, gamma, beta, x0);
    gather_kernel<<<(NSTEP * BSZ * EE) / 256, 256, 0, stream>>>(x0, emb, gt, X);
    // 124 m-groups (MT=2) x 64 n-groups = 7936 waves -> 992 blocks
    gatesx_kernel<<<((NSTEP * BSZ / 32) * ((4 * HH) / 64)) / 8, 256, 0, stream>>>(X, W_ih, b_ih, b_hh, G);
    zero_kernel  <<<(2 * BSZ * HH) / 256, 256, 0, stream>>>(hbuf); // zeroes h and c

    for (int t = 0; t < NSTEP; ++t) {
        // 8 mtiles x 128 n-groups = 1024 waves per split -> grid (128, 2)
        gatesh_kernel<<<dim3(((BSZ / 16) * ((4 * HH) / 32)) / 8, 2), 256, 0, stream>>>(
            hbuf, W_hh, G, t, gates);
        lstm_update_kernel<<<(BSZ * HH) / 256, 256, 0, stream>>>(gates, hbuf, cbuf, Hall, t);
    }

    // 120 m-groups (MT=2) x 125 n-groups (NT=5) = 15000 waves -> 1875 blocks
    fc_kernel<<<((TMAXX * BSZ / 32) * (VV / 80)) / 8, 256, 0, stream>>>(Hall, W_fc, b_fc, clen, out);
    (void)in_sizes; (void)n_in; (void)out_size; (void)ws_size;
}